// DN1_90864328114189
// MI455X (gfx1250) — compile-verified
//
#include <hip/hip_runtime.h>
#include <math.h>

typedef __attribute__((ext_vector_type(2))) float v2f;
typedef __attribute__((ext_vector_type(8))) float v8f;
typedef __attribute__((ext_vector_type(4))) unsigned int u32x4;
typedef __attribute__((ext_vector_type(4))) int i32x4;
typedef __attribute__((ext_vector_type(8))) int i32x8;

#define X_SIZE 2048
#define NUM_NEURONS 4096
#define Z_SIZE 64
#define DDIM 6208              // X_SIZE + NUM_NEURONS + Z_SIZE
#define Y0 2048
#define Z0 6144
#define TOPK 16
#define ZTOPK 8
#define EPS_TIE 1e-9f
#define RND 0.5f
#define NEG_INF (-3.402823466e+38f)

// Tensor Data Mover availability (device pass only; host pass falls back)
#if __has_builtin(__builtin_amdgcn_tensor_load_to_lds) && __has_builtin(__builtin_amdgcn_s_wait_tensorcnt)
#define CDNA5_HAVE_TDM 1
#else
#define CDNA5_HAVE_TDM 0
#endif

// ---------------------------------------------------------------------------
// 1) Build inpt = concat(x, y_response, z)
// ---------------------------------------------------------------------------
__global__ void k_build_inpt(const float* __restrict__ x,
                             const float* __restrict__ y,
                             const float* __restrict__ z,
                             float* __restrict__ inpt) {
  int i = blockIdx.x * blockDim.x + threadIdx.x;
  if (i >= DDIM) return;
  float v;
  if (i < X_SIZE)      v = x[i];
  else if (i < Z0)     v = y[i - X_SIZE];
  else                 v = z[i - Z0];
  inpt[i] = v;
}

// ---------------------------------------------------------------------------
// 2) GEMV via V_WMMA_F32_16X16X4_F32.
//    Block = 256 threads = 8 waves; block handles a 16-row tile of neurons.
//    Wave w accumulates K-slice [w*776, (w+1)*776) into C column 0.
//    A layout (ISA 7.12.2, 32-bit A 16x4): lanes 0-15 -> M=lane, K = k,k+1;
//    lanes 16-31 -> M=lane-16, K = k+2,k+3.  B (4x16) mirrors: lane half
//    selects K pair, N = lane&15; we only populate column N==0.
//    C/D: lane 0 holds rows 0..7 in VGPRs 0..7, lane 16 holds rows 8..15.
//    inpt (24.8 KB) is staged into LDS by the Tensor Data Mover: one DMA
//    issued by wave 0, completion via s_wait_tensorcnt, then wg barrier.
// ---------------------------------------------------------------------------
struct GemvSmem {
  float in[DDIM];        // must be at LDS offset 0 (sole shared aggregate)
  float part[8][16];
};

__global__ void k_gemv_wmma(const float* __restrict__ neurons,
                            const float* __restrict__ inpt,
                            float* __restrict__ response) {
  __shared__ GemvSmem sm;

#if CDNA5_HAVE_TDM
  if (threadIdx.x < 32) {
    // D# group 0: count=1 | lds_addr=0 | global_addr | type=2
    unsigned long long ga = (unsigned long long)(uintptr_t)inpt;
    u32x4 g0;
    g0[0] = 1u;                                   // count = 1 (valid D#)
    g0[1] = 0u;                                   // lds_addr: sm.in @ offset 0
    g0[2] = (unsigned)ga;                         // global_addr[31:0]
    g0[3] = (unsigned)((ga >> 32) & 0x01FFFFFFull) | (2u << 30); // [56:32]|type=2
    // D# group 1: 1-row 2D tile, data_size=4B, dim0 = tile0 = stride0 = DDIM
    i32x8 g1;
    g1[0] = 0x00020000;                           // wg_mask=0, data_size=2 (4B)
    g1[1] = (int)(DDIM << 16);                    // tensor_dim0[15:0] << 16
    g1[2] = 0x00010000;                           // tensor_dim1 = 1
    g1[3] = (int)(DDIM << 16);                    // tile_dim0 = DDIM
    g1[4] = 1;                                    // tile_dim1 = 1
    g1[5] = DDIM;                                 // tensor_dim0_stride[31:0]
    g1[6] = (int)(DDIM << 16);                    // tensor_dim1_stride[15:0]
    g1[7] = 0;
    i32x4 gz = {0, 0, 0, 0};
#if __clang_major__ >= 23
    i32x8 gz8 = {0, 0, 0, 0, 0, 0, 0, 0};
    __builtin_amdgcn_tensor_load_to_lds(g0, g1, gz, gz, gz8, 0);
#else
    __builtin_amdgcn_tensor_load_to_lds(g0, g1, gz, gz, 0);
#endif
    __builtin_amdgcn_s_wait_tensorcnt(0);
  }
  __syncthreads();
#else
  for (int i = threadIdx.x; i < DDIM; i += 256) sm.in[i] = inpt[i];
  __syncthreads();
#endif

  const int wave = threadIdx.x >> 5;
  const int lane = threadIdx.x & 31;
  const int r    = lane & 15;      // row-in-tile (A) / column N (B)
  const int hi   = lane >> 4;      // 0: K pair {0,1}; 1: K pair {2,3}
  const int koff = hi * 2;
  const int tile = blockIdx.x;

  const float* row = neurons + (size_t)(tile * 16 + r) * DDIM;
  const int kbeg = wave * (DDIM / 8);       // 776 columns per wave
  const int kend = kbeg + (DDIM / 8);

  v8f acc = {};
  for (int k = kbeg; k < kend; k += 4) {
    v2f a = *reinterpret_cast<const v2f*>(row + k + koff);   // 8B-aligned
    float b0 = sm.in[k + koff];
    float b1 = sm.in[k + koff + 1];
    v2f b;
    b.x = (r == 0) ? b0 : 0.0f;   // only B column 0 is live
    b.y = (r == 0) ? b1 : 0.0f;
    acc = __builtin_amdgcn_wmma_f32_16x16x4_f32(
        /*neg_a=*/false, a, /*neg_b=*/false, b,
        /*c_mod=*/(short)0, acc, /*reuse_a=*/false, /*reuse_b=*/false);
  }

  if (r == 0) {
#pragma unroll
    for (int m = 0; m < 8; ++m) sm.part[wave][hi * 8 + m] = acc[m];
  }
  __syncthreads();

  if (threadIdx.x < 16) {
    float s = 0.0f;
#pragma unroll
    for (int w = 0; w < 8; ++w) s += sm.part[w][threadIdx.x];
    response[tile * 16 + threadIdx.x] = s;
  }
}

// ---------------------------------------------------------------------------
// 3) Selection / scalar pass (single block, 256 threads):
//    - z top-9, y top-17 (iterative max, smaller-index tiebreak = JAX order)
//    - x max + all-zero flag
//    - writes final[], z_response[], and per-row (upd_val, upd_has)
// ---------------------------------------------------------------------------
__global__ void k_select(const float* __restrict__ resp,
                         float* __restrict__ finalv,   // length DDIM
                         float* __restrict__ zresp,    // length 64
                         float* __restrict__ upd_val,
                         float* __restrict__ upd_has) {
  __shared__ float s_y[NUM_NEURONS];
  __shared__ float s_v[256];
  __shared__ int   s_i[256];
  __shared__ float s_tv[TOPK + 1];
  __shared__ int   s_ti[TOPK + 1];
  __shared__ float s_z[Z_SIZE];
  __shared__ float s_zv[ZTOPK + 1];
  __shared__ int   s_zi[ZTOPK + 1];

  const int t = threadIdx.x;

  for (int i = t; i < DDIM; i += 256) {
    finalv[i]  = 0.0f;
    upd_val[i] = 0.0f;
    upd_has[i] = 0.0f;
  }
  for (int i = t; i < NUM_NEURONS; i += 256) s_y[i] = resp[Y0 + i];
  if (t < Z_SIZE) s_z[t] = resp[Z0 + t];
  __syncthreads();

  // ---- z region: top-9 of 64 values (serial on thread 0, trivial size)
  if (t == 0) {
    for (int s = 0; s <= ZTOPK; ++s) {
      float bv = NEG_INF; int bi = 0;
      for (int i = 0; i < Z_SIZE; ++i) {
        float v = s_z[i];
        if (v > bv) { bv = v; bi = i; }
      }
      s_zv[s] = bv; s_zi[s] = bi; s_z[bi] = NEG_INF;
    }
  }
  __syncthreads();

  // ---- x region: max + any-nonzero
  float lm = NEG_INF;
  int any_nz = 0;
  for (int i = t; i < X_SIZE; i += 256) {
    float v = resp[i];
    lm = fmaxf(lm, v);
    if (v != 0.0f) any_nz = 1;
  }
  s_v[t] = lm; s_i[t] = any_nz;
  __syncthreads();
  for (int off = 128; off > 0; off >>= 1) {
    if (t < off) { s_v[t] = fmaxf(s_v[t], s_v[t + off]); s_i[t] |= s_i[t + off]; }
    __syncthreads();
  }
  const float mx = s_v[0];
  const float tx = (s_i[0] == 0) ? 1.0f : 0.0f;
  const float denx = mx + EPS_TIE * tx * RND;
  __syncthreads();

  // ---- y region: top-17 of 4096 via 17 iterative block max-reductions
  for (int s = 0; s <= TOPK; ++s) {
    float bv = NEG_INF; int bi = 0x7fffffff;
    for (int i = t; i < NUM_NEURONS; i += 256) {
      float v = s_y[i];
      if (v > bv || (v == bv && i < bi)) { bv = v; bi = i; }
    }
    s_v[t] = bv; s_i[t] = bi;
    __syncthreads();
    for (int off = 128; off > 0; off >>= 1) {
      if (t < off) {
        float v2 = s_v[t + off]; int i2 = s_i[t + off];
        if (v2 > s_v[t] || (v2 == s_v[t] && i2 < s_i[t])) { s_v[t] = v2; s_i[t] = i2; }
      }
      __syncthreads();
    }
    if (t == 0) { s_tv[s] = s_v[0]; s_ti[s] = s_i[0]; s_y[s_i[0]] = NEG_INF; }
    __syncthreads();
  }

  // ---- write x outputs
  for (int i = t; i < X_SIZE; i += 256) {
    float xv = resp[i] / denx;
    finalv[i] = xv;
    if (i < X_SIZE - 1) { upd_val[i] = xv; upd_has[i] = 1.0f; }
  }

  // ---- write z / y top-k outputs
  if (t == 0) {
    const float zlast = s_zv[ZTOPK];
    const float dz = s_zv[0] - zlast;
    for (int j = 0; j < ZTOPK; ++j) {
      int gi = Z0 + s_zi[j];
      finalv[gi]  = (s_zv[j] - zlast) / dz;
      upd_val[gi] = s_zv[j];
      upd_has[gi] = 1.0f;
    }
    const float ylast = s_tv[TOPK];
    float ty = 0.0f;
    for (int j = 0; j < TOPK; ++j) if (s_tv[j] == ylast) ty = 1.0f;
    const float dy = s_tv[0] - ylast + EPS_TIE * ty * RND;
    for (int j = 0; j < TOPK; ++j) {
      int gi = Y0 + s_ti[j];
      finalv[gi]  = (s_tv[j] - ylast) / dy;
      upd_val[gi] = s_tv[j];
      upd_has[gi] = 1.0f;
    }
  }
  __syncthreads();

  if (t < Z_SIZE) zresp[t] = finalv[Z0 + t];
}

// ---------------------------------------------------------------------------
// 4) Row update / copy: one block per row. Updated rows:
//    new = ((a-1)/a)*row + (v/a)*inpt, normalized; else plain copy.
//    float4 streaming, new row kept in registers between the two phases.
// ---------------------------------------------------------------------------
__global__ void k_update(const float* __restrict__ neurons,
                         const float* __restrict__ ages,
                         const float* __restrict__ inpt,
                         const float* __restrict__ upd_val,
                         const float* __restrict__ upd_has,
                         float* __restrict__ out_neurons,
                         float* __restrict__ out_ages) {
  const int d = blockIdx.x;
  const int t = threadIdx.x;
  const float a = ages[d];
  const float* row = neurons + (size_t)d * DDIM;
  float* orow = out_neurons + (size_t)d * DDIM;
  const int n4 = DDIM / 4;   // 1552 float4s per row

  if (upd_has[d] == 0.0f) {
    const float4* r4 = reinterpret_cast<const float4*>(row);
    float4* o4 = reinterpret_cast<float4*>(orow);
    for (int i = t; i < n4; i += 256) o4[i] = r4[i];
    if (t == 0) out_ages[d] = a;
    return;
  }

  const float c0 = (a - 1.0f) / a;
  const float c1 = upd_val[d] / a;
  const float4* r4 = reinterpret_cast<const float4*>(row);
  const float4* i4 = reinterpret_cast<const float4*>(inpt);

  float4 v[7];
  float ss = 0.0f;
#pragma unroll
  for (int j = 0; j < 7; ++j) {
    int i = t + j * 256;
    if (i < n4) {
      float4 rv = r4[i];
      float4 iv = i4[i];
      float4 nv;
      nv.x = c0 * rv.x + c1 * iv.x;
      nv.y = c0 * rv.y + c1 * iv.y;
      nv.z = c0 * rv.z + c1 * iv.z;
      nv.w = c0 * rv.w + c1 * iv.w;
      v[j] = nv;
      ss += nv.x * nv.x + nv.y * nv.y + nv.z * nv.z + nv.w * nv.w;
    }
  }

  __shared__ float s_red[256];
  s_red[t] = ss;
  __syncthreads();
  for (int off = 128; off > 0; off >>= 1) {
    if (t < off) s_red[t] += s_red[t + off];
    __syncthreads();
  }
  const float inv = 1.0f / (sqrtf(s_red[0]) + 1e-12f);

  float4* o4 = reinterpret_cast<float4*>(orow);
#pragma unroll
  for (int j = 0; j < 7; ++j) {
    int i = t + j * 256;
    if (i < n4) {
      float4 nv = v[j];
      nv.x *= inv; nv.y *= inv; nv.z *= inv; nv.w *= inv;
      o4[i] = nv;
    }
  }
  if (t == 0) out_ages[d] = a + 1.0f;
}

// ---------------------------------------------------------------------------
extern "C" void kernel_launch(void* const* d_in, const int* in_sizes, int n_in,
                              void* d_out, int out_size, void* d_ws, size_t ws_size,
                              hipStream_t stream) {
  // setup_inputs order: x, z, y_response, neurons, ages
  const float* x       = (const float*)d_in[0];
  const float* z       = (const float*)d_in[1];
  const float* y_resp  = (const float*)d_in[2];
  const float* neurons = (const float*)d_in[3];
  const float* ages    = (const float*)d_in[4];

  float* out    = (float*)d_out;
  float* zresp  = out;                                   // 64
  float* finalv = out + Z_SIZE;                          // 6208
  float* oneur  = finalv + DDIM;                         // 6208*6208
  float* oages  = oneur + (size_t)DDIM * DDIM;           // 6208

  float* ws      = (float*)d_ws;
  float* inpt    = ws;                                   // 6208
  float* resp    = ws + DDIM;                            // 6208
  float* upd_val = ws + 2 * DDIM;                        // 6208
  float* upd_has = ws + 3 * DDIM;                        // 6208

  k_build_inpt<<<(DDIM + 255) / 256, 256, 0, stream>>>(x, y_resp, z, inpt);
  k_gemv_wmma<<<DDIM / 16, 256, 0, stream>>>(neurons, inpt, resp);
  k_select<<<1, 256, 0, stream>>>(resp, finalv, zresp, upd_val, upd_has);
  k_update<<<DDIM, 256, 0, stream>>>(neurons, ages, inpt, upd_val, upd_has,
                                     oneur, oages);
}